// QuantumNATEnhanced_65481071403804
// MI455X (gfx1250) — compile-verified
//
#include <hip/hip_runtime.h>

// ---------------------------------------------------------------------------
// QuantumNAT: analytic collapse of the circuit.
//   theta_i = mean(x[b, 0, 6i:6i+6, 0:6]),  c_i = cos(theta_i)
//   z = { c1*c2*c3, c0*c1, c0*c1*c2, c0*c1*c2*c3 }   then batchnorm over B.
// Phase 1 streams the 205MB input (memory bound, ~9us floor @ 23.3 TB/s) and
// performs the pooling contraction with V_WMMA_F32_16X16X4_F32:
//   D(16 imgs x 16) += A(16 imgs x 4 rowsums) x B(4 x 16 selection/36)
// six K-chunks cover the 24 rows. Deterministic fixed-order reductions only.
// ---------------------------------------------------------------------------

typedef float v2f __attribute__((ext_vector_type(2)));
typedef float v8f __attribute__((ext_vector_type(8)));

#define BATCH            65536
#define IMG_STRIDE       784      // 28*28 floats
#define ROW_STRIDE       28
#define IMGS_PER_WAVE    16
#define WAVES_PER_BLOCK  8
#define IMGS_PER_BLOCK   (IMGS_PER_WAVE * WAVES_PER_BLOCK)   // 128
#define NBLOCKS          (BATCH / IMGS_PER_BLOCK)            // 512
#define EPS_F            1e-5f

// ---------------- Phase 1: pool (via WMMA) + cos + products + partials ------
__global__ __launch_bounds__(256) void qnat_phase1(const float* __restrict__ x,
                                                   float* __restrict__ raw_out,
                                                   float* __restrict__ partials)
{
    __shared__ float lds_cos[WAVES_PER_BLOCK][IMGS_PER_WAVE][4];
    __shared__ float lds_red[IMGS_PER_BLOCK][8];

    const int tid  = threadIdx.x;
    const int wave = tid >> 5;
    const int lane = tid & 31;
    const int m    = lane & 15;   // image slot (A: M index) and B: N index
    const int hi   = lane >> 4;   // 0 -> K0/K1 rows of chunk, 1 -> K2/K3

    const int g0 = blockIdx.x * IMGS_PER_BLOCK + wave * IMGS_PER_WAVE;
    const float* img = x + (size_t)(g0 + m) * IMG_STRIDE;

    v8f acc = {};
    const float w = 1.0f / 36.0f;

    // 24 rows = 6 K-chunks of 4. Lane holds rowsums for rows 4k+2*hi, +1.
    // B[K][N] = 1/36 if N == row/6 (N<4 implicit), giving D[M][N] = theta_N.
#pragma unroll
    for (int k = 0; k < 6; ++k) {
        const int r0 = 4 * k + 2 * hi;
        const float* p0 = img + r0 * ROW_STRIDE;
        const float* p1 = p0 + ROW_STRIDE;
        float4 a0 = *(const float4*)(p0);
        float2 b0 = *(const float2*)(p0 + 4);
        float4 a1 = *(const float4*)(p1);
        float2 b1 = *(const float2*)(p1 + 4);

        v2f A;
        A.x = a0.x + a0.y + a0.z + a0.w + b0.x + b0.y;
        A.y = a1.x + a1.y + a1.z + a1.w + b1.x + b1.y;

        v2f Bm;
        Bm.x = (m == (r0    ) / 6) ? w : 0.0f;
        Bm.y = (m == (r0 + 1) / 6) ? w : 0.0f;

        acc = __builtin_amdgcn_wmma_f32_16x16x4_f32(
                  false, A, false, Bm, (short)0, acc, false, false);
    }

    // D layout: VGPR j -> image (hi ? j+8 : j), column N = m. theta_N at N<4.
    if (m < 4) {
        const int imgbase = hi * 8;
#pragma unroll
        for (int j = 0; j < 8; ++j)
            lds_cos[wave][imgbase + j][m] = __cosf(acc[j]);
    }
    __syncthreads();

    if (lane < 16) {
        const int b = g0 + lane;
        const float c0 = lds_cos[wave][lane][0];
        const float c1 = lds_cos[wave][lane][1];
        const float c2 = lds_cos[wave][lane][2];
        const float c3 = lds_cos[wave][lane][3];
        float4 z;
        z.x = c1 * c2 * c3;        // z0
        z.y = c0 * c1;             // z1
        z.z = z.y * c2;            // z2
        z.w = z.z * c3;            // z3
        *(float4*)(raw_out + (size_t)b * 4) = z;

        const int slot = wave * IMGS_PER_WAVE + lane;
        lds_red[slot][0] = z.x;       lds_red[slot][1] = z.y;
        lds_red[slot][2] = z.z;       lds_red[slot][3] = z.w;
        lds_red[slot][4] = z.x * z.x; lds_red[slot][5] = z.y * z.y;
        lds_red[slot][6] = z.z * z.z; lds_red[slot][7] = z.w * z.w;
    }
    __syncthreads();

    // Fixed-order (deterministic) per-channel block reduction.
    if (tid < 8) {
        float s = 0.0f;
        for (int i = 0; i < IMGS_PER_BLOCK; ++i) s += lds_red[i][tid];
        partials[blockIdx.x * 8 + tid] = s;
    }
}

// ---------------- Phase 2: global stats (single block, fixed order) ---------
__global__ void qnat_phase2(const float* __restrict__ partials,
                            float* __restrict__ stats)
{
    __shared__ float sq[8];
    const int t = threadIdx.x;
    if (t < 8) {
        float s = 0.0f;
        for (int i = 0; i < NBLOCKS; ++i) s += partials[i * 8 + t];
        sq[t] = s;
    }
    __syncthreads();
    if (t < 4) {
        const float mean = sq[t] * (1.0f / (float)BATCH);
        const float var  = sq[t + 4] * (1.0f / (float)BATCH) - mean * mean;
        stats[t]     = mean;
        stats[t + 4] = rsqrtf(var + EPS_F);
    }
}

// ---------------- Phase 3: in-place normalize of d_out ----------------------
__global__ __launch_bounds__(256) void qnat_phase3(float* __restrict__ out,
                                                   const float* __restrict__ stats,
                                                   const float* __restrict__ gamma,
                                                   const float* __restrict__ beta)
{
    const int idx = blockIdx.x * blockDim.x + threadIdx.x;
    const int j = idx & 3;
    const float mean = stats[j];
    const float inv  = stats[j + 4];
    const float v = out[idx];
    out[idx] = gamma[j] * (v - mean) * inv + beta[j];
}

// ---------------------------------------------------------------------------
extern "C" void kernel_launch(void* const* d_in, const int* in_sizes, int n_in,
                              void* d_out, int out_size, void* d_ws, size_t ws_size,
                              hipStream_t stream)
{
    const float* x     = (const float*)d_in[0];
    const float* gamma = (const float*)d_in[1];
    const float* beta  = (const float*)d_in[2];
    float* out = (float*)d_out;

    float* partials = (float*)d_ws;             // NBLOCKS * 8 floats
    float* stats    = partials + NBLOCKS * 8;   // 8 floats

    qnat_phase1<<<NBLOCKS, 256, 0, stream>>>(x, out, partials);
    qnat_phase2<<<1, 32, 0, stream>>>(partials, stats);
    qnat_phase3<<<(BATCH * 4) / 256, 256, 0, stream>>>(out, stats, gamma, beta);
}